// TensorPCA_11458972745942
// MI455X (gfx1250) — compile-verified
//
#include <hip/hip_runtime.h>
#include <math.h>

typedef __attribute__((ext_vector_type(2))) float v2f;
typedef __attribute__((ext_vector_type(8))) float v8f;

#define RANK    64
#define NUMVOX  64
#define FEAT    192
#define FLATN   786432          // FEAT * 16^3
#define VOLELTS ((size_t)64*64*64*192)

// ---------------------------------------------------------------------------
// Kernel 1: A' = U * exp(S)  (64x64, row-major, k fastest)
// ---------------------------------------------------------------------------
__global__ void scale_U_kernel(const float* __restrict__ U,
                               const float* __restrict__ S,
                               float* __restrict__ Ap) {
    int t = blockIdx.x * blockDim.x + threadIdx.x;
    if (t < NUMVOX * RANK) {
        int k = t & (RANK - 1);
        Ap[t] = U[t] * __expf(S[k]);
    }
}

// ---------------------------------------------------------------------------
// Kernel 2: recon = A' @ Vh via V_WMMA_F32_16X16X4_F32, scattered directly
// into the merged volume with channel-innermost layout vol[X][Y][Z][C].
//   recon[m, n]: m = bx*16+by*4+bz (voxel block), n = c*4096 + i0*256 + i1*16 + i2
//   volume coord: X = bx*16+i0, Y = by*16+i1, Z = bz*16+i2
// Block = 256 threads = 8 waves; wave -> (rowTile = w&3, colTile = 2*bid + w>>2)
// ---------------------------------------------------------------------------
__global__ void __launch_bounds__(256) pca_gemm_wmma(const float* __restrict__ Ap,
                                                     const float* __restrict__ Vh,
                                                     float* __restrict__ vol) {
    const int lane    = threadIdx.x & 31;
    const int w       = threadIdx.x >> 5;
    const int rowTile = w & 3;                        // Mbase = rowTile*16 (== bx)
    const int colTile = blockIdx.x * 2 + (w >> 2);
    const int lm      = lane & 15;
    const int hi      = lane >> 4;                    // 0: lanes 0-15, 1: lanes 16-31
    const int khalf   = hi * 2;                       // K pair (0,1) or (2,3)

    const long   n    = (long)colTile * 16 + lm;      // this lane's N column
    const float* Arow = Ap + (rowTile * 16 + lm) * RANK;

    v8f acc = {0.f, 0.f, 0.f, 0.f, 0.f, 0.f, 0.f, 0.f};

#pragma unroll
    for (int kk = 0; kk < RANK; kk += 4) {
        v2f a, b;
        // A fragment: row M = Mbase+lm, K = kk+khalf, kk+khalf+1
        a.x = Arow[kk + khalf];
        a.y = Arow[kk + khalf + 1];
        // B fragment: col N = n, K = kk+khalf, kk+khalf+1  (Vh row-major, stride FLATN)
        const float* Bp = Vh + (size_t)(kk + khalf) * FLATN + n;
        b.x = Bp[0];
        b.y = Bp[FLATN];
        acc = __builtin_amdgcn_wmma_f32_16x16x4_f32(
            /*neg_a=*/false, a, /*neg_b=*/false, b,
            /*c_mod=*/(short)0, acc, /*reuse_a=*/false, /*reuse_b=*/false);
    }

    // Decode this lane's column n -> (c, i0, i1, i2); fixed for all 8 D elements.
    const int c   = (int)(n >> 12);
    const int rem = (int)(n & 4095);
    const int i0  = rem >> 8;
    const int i1  = (rem >> 4) & 15;
    const int i2  = rem & 15;

    // vol[X][Y][Z][C]: idx = ((X*64 + Y)*64 + Z)*192 + c
    // X = rowTile*16 + i0 fixed; per acc element only (by,bz) of m vary.
    const size_t base = (((size_t)(rowTile * 16 + i0) * 4096)
                         + (size_t)i1 * 64 + (size_t)i2) * FEAT + c;

#pragma unroll
    for (int v = 0; v < 8; ++v) {
        const int mlow = v + hi * 8;                  // m & 15
        const int by   = mlow >> 2;
        const int bz   = mlow & 3;
        vol[base + (size_t)by * (16 * 64 * 192) + (size_t)bz * (16 * 192)] = acc[v];
    }
}

// ---------------------------------------------------------------------------
// Kernel 3: trilinear sampling. 192 threads/block = one channel per thread;
// each corner fetch is a contiguous 768B burst; output fully coalesced.
// Axes: pts (x,y,z) -> (W=58, H=62, D=60); vol idx = ((zD*64 + yH)*64 + xW)*192 + c
// ---------------------------------------------------------------------------
__global__ void __launch_bounds__(192) pca_sample(const float* __restrict__ vol,
                                                  const float* __restrict__ xyz,
                                                  float* __restrict__ out,
                                                  int npts) {
    const int c = threadIdx.x;   // 0..191
    for (int p = blockIdx.x; p < npts; p += gridDim.x) {
        const float x = xyz[3 * p + 0];
        const float y = xyz[3 * p + 1];
        const float z = xyz[3 * p + 2];

        const float fx = (x + 1.0f) * 0.5f * 57.0f;   // W-1 = 57
        const float fy = (y + 1.0f) * 0.5f * 61.0f;   // H-1 = 61
        const float fz = (z + 1.0f) * 0.5f * 59.0f;   // D-1 = 59

        const float x0f = floorf(fx), y0f = floorf(fy), z0f = floorf(fz);
        const float wx = fx - x0f, wy = fy - y0f, wz = fz - z0f;
        const float ox = 1.0f - wx, oy = 1.0f - wy, oz = 1.0f - wz;

        int x0 = min(max((int)x0f, 0), 57);
        int y0 = min(max((int)y0f, 0), 61);
        int z0 = min(max((int)z0f, 0), 59);
        const int x1 = min(x0 + 1, 57);
        const int y1 = min(y0 + 1, 61);
        const int z1 = min(z0 + 1, 59);

        const size_t z0b = (size_t)z0 * 4096, z1b = (size_t)z1 * 4096;
        const size_t y0b = (size_t)y0 * 64,   y1b = (size_t)y1 * 64;

        float r;
        r  = vol[(z0b + y0b + x0) * FEAT + c] * (oz * oy * ox);
        r += vol[(z0b + y0b + x1) * FEAT + c] * (oz * oy * wx);
        r += vol[(z0b + y1b + x0) * FEAT + c] * (oz * wy * ox);
        r += vol[(z0b + y1b + x1) * FEAT + c] * (oz * wy * wx);
        r += vol[(z1b + y0b + x0) * FEAT + c] * (wz * oy * ox);
        r += vol[(z1b + y0b + x1) * FEAT + c] * (wz * oy * wx);
        r += vol[(z1b + y1b + x0) * FEAT + c] * (wz * wy * ox);
        r += vol[(z1b + y1b + x1) * FEAT + c] * (wz * wy * wx);

        out[(size_t)p * FEAT + c] = r;
    }
}

// ---------------------------------------------------------------------------
// Launch: inputs in setup_inputs() order: xyz_sampled, U, S, Vh (all f32).
// Workspace layout: [0 .. 4096) floats = A', then 64^3*192 floats = volume.
// ---------------------------------------------------------------------------
extern "C" void kernel_launch(void* const* d_in, const int* in_sizes, int n_in,
                              void* d_out, int out_size, void* d_ws, size_t ws_size,
                              hipStream_t stream) {
    const float* xyz = (const float*)d_in[0];
    const float* U   = (const float*)d_in[1];
    const float* S   = (const float*)d_in[2];
    const float* Vh  = (const float*)d_in[3];
    float*       out = (float*)d_out;

    float* Ap  = (float*)d_ws;
    float* vol = Ap + NUMVOX * RANK;

    const int npts = in_sizes[0] / 3;

    // 1) A' = U * exp(S)
    scale_U_kernel<<<(NUMVOX * RANK + 255) / 256, 256, 0, stream>>>(U, S, Ap);

    // 2) WMMA GEMM fused with block-merge into vol[X][Y][Z][C]
    //    49152 column tiles / 2 per block = 24576 blocks of 8 waves
    pca_gemm_wmma<<<FLATN / 16 / 2, 256, 0, stream>>>(Ap, Vh, vol);

    // 3) Trilinear sampling, one channel per thread
    pca_sample<<<4096, FEAT, 0, stream>>>(vol, xyz, out, npts);
}